// HeatmapOffsetmapLoss_65034394796079
// MI455X (gfx1250) — compile-verified
//
#include <hip/hip_runtime.h>
#include <hip/hip_bf16.h>
#include <stdint.h>

#define NPAR 19
#define C3 57
#define PAIRS 152            // 8 * 19
#define HW 512
#define NPIX (HW * HW)       // 262144
#define NSLICE 32
#define SLICE_FLOATS (NPIX / NSLICE)   // 8192 floats per block
#define WAVE_FLOATS (SLICE_FLOATS / 8) // 1024 floats per wave
#define ITERS 8                        // 128 floats per async op per wave
#define RAD2 1600.0f

// ---------------- CDNA5 async-to-LDS helpers (gfx1250) ----------------
__device__ __forceinline__ void async_issue_b128(unsigned ldsAddr, unsigned gOff,
                                                 const float* base) {
  // GVS mode: mem_addr = SGPR64(base) + VGPR32(gOff); VDST = LDS byte address.
  asm volatile("global_load_async_to_lds_b128 %0, %1, %2"
               :: "v"(ldsAddr), "v"(gOff), "s"(base)
               : "memory");
}
template <int N>
__device__ __forceinline__ void wait_asynccnt() {
  asm volatile("s_wait_asynccnt %0" :: "i"(N) : "memory");
}
__device__ __forceinline__ void wait_dscnt0() {
  asm volatile("s_wait_dscnt 0" ::: "memory");
}
// Low 32 bits of a flat pointer to __shared__ == AS3 (LDS) byte offset.
__device__ __forceinline__ unsigned lds_addr(const void* p) {
  return (unsigned)(uintptr_t)p;
}

// ---------------- Kernel A: streamed BCE over fm_p ----------------
// partial[(pid*NSLICE + slice)*2 + {0,1}] = { sum(max(z,0)+log1p(exp(-|z|))), sum(z*heat) }
__global__ __launch_bounds__(256) void bce_stream(const float* __restrict__ fm,
                                                  const float* __restrict__ lm,
                                                  float* __restrict__ partial) {
  __shared__ __align__(16) float sbuf[8][2][128];  // per-wave double buffer, 8 KB
  __shared__ float red[2][8];

  const int pid   = blockIdx.x;       // 0..151
  const int slice = blockIdx.y;       // 0..31
  const int b = pid / NPAR, p = pid % NPAR;

  const float x = floorf(lm[(b * NPAR + p) * 2 + 0] * (float)HW);
  const float y = floorf(lm[(b * NPAR + p) * 2 + 1] * (float)HW);

  const float* base = fm + (size_t)(b * C3 + p) * NPIX;
  const int wave = threadIdx.x >> 5;
  const int lane = threadIdx.x & 31;

  const int chunkBase = slice * SLICE_FLOATS + wave * WAVE_FLOATS;
  const unsigned ldsA = lds_addr(&sbuf[wave][0][lane * 4]);
  const unsigned ldsB = lds_addr(&sbuf[wave][1][lane * 4]);

  // prime the pipeline: iteration 0 -> buffer A
  async_issue_b128(ldsA, (unsigned)((chunkBase + lane * 4) * 4), base);

  float sMain = 0.0f, sZH = 0.0f;

#pragma unroll
  for (int k = 0; k < ITERS; ++k) {
    if (k + 1 < ITERS) {
      wait_dscnt0();  // previous reads of the buffer we are about to overwrite are done
      const unsigned lds = ((k + 1) & 1) ? ldsB : ldsA;
      async_issue_b128(lds, (unsigned)((chunkBase + (k + 1) * 128 + lane * 4) * 4), base);
      wait_asynccnt<1>();  // in-order completion: load k has landed in LDS
    } else {
      wait_asynccnt<0>();
    }

    const float* bufp = (k & 1) ? &sbuf[wave][1][lane * 4] : &sbuf[wave][0][lane * 4];
    const float4 v = *(const float4*)bufp;

    const int idx = chunkBase + k * 128 + lane * 4;  // multiple of 4 -> stays in one row
    const float fi = (float)(idx >> 9);
    const int j0 = idx & (HW - 1);
    const float dx = x - fi;
    const float dxx = dx * dx;

    const float zs[4] = {v.x, v.y, v.z, v.w};
#pragma unroll
    for (int c = 0; c < 4; ++c) {
      const float z = zs[c];
      const float dy = y - (float)(j0 + c);
      const bool inside = (dxx + dy * dy) <= RAD2;
      const float az = fabsf(z);
      const float sp = __logf(1.0f + __expf(-az));  // log1p(exp(-|z|))
      sMain += fmaxf(z, 0.0f) + sp;
      if (inside) sZH += z;
    }
  }

  // wave32 reduction, then cross-wave via LDS
#pragma unroll
  for (int off = 16; off > 0; off >>= 1) {
    sMain += __shfl_down(sMain, off, 32);
    sZH   += __shfl_down(sZH, off, 32);
  }
  if (lane == 0) { red[0][wave] = sMain; red[1][wave] = sZH; }
  __syncthreads();
  if (threadIdx.x == 0) {
    float m = 0.0f, zh = 0.0f;
#pragma unroll
    for (int w = 0; w < 8; ++w) { m += red[0][w]; zh += red[1][w]; }
    partial[(pid * NSLICE + slice) * 2 + 0] = m;
    partial[(pid * NSLICE + slice) * 2 + 1] = zh;
  }
}

// ---------------- Kernel B: masked L1 over the <=81x81 disc only ----------------
// res[pid*3 + {0,1,2}] = { cnt, sum|fm_x - dx/40|*heat, sum|fm_y - dy/40|*heat }
__global__ __launch_bounds__(256) void l1_disc(const float* __restrict__ fm,
                                               const float* __restrict__ lm,
                                               float* __restrict__ res) {
  __shared__ float red[3][8];
  const int pid = blockIdx.x;
  const int b = pid / NPAR, p = pid % NPAR;

  const float x = floorf(lm[(b * NPAR + p) * 2 + 0] * (float)HW);
  const float y = floorf(lm[(b * NPAR + p) * 2 + 1] * (float)HW);
  const int xi = (int)x, yi = (int)y;

  const float* fmx = fm + (size_t)(b * C3 + NPAR + p) * NPIX;
  const float* fmy = fm + (size_t)(b * C3 + 2 * NPAR + p) * NPIX;

  const int i0 = max(0, xi - 40), i1 = min(HW - 1, xi + 40);
  const int j0 = max(0, yi - 40), j1 = min(HW - 1, yi + 40);
  const int nj = j1 - j0 + 1;
  const int tot = (i1 - i0 + 1) * nj;

  float cnt = 0.0f, sx = 0.0f, sy = 0.0f;
  for (int t = threadIdx.x; t < tot; t += 256) {
    const int i = i0 + t / nj;
    const int j = j0 + t % nj;
    const float dx = x - (float)i;
    const float dy = y - (float)j;
    if (dx * dx + dy * dy <= RAD2) {
      cnt += 1.0f;
      sx += fabsf(fmx[i * HW + j] - dx * (1.0f / 40.0f));
      sy += fabsf(fmy[i * HW + j] - dy * (1.0f / 40.0f));
    }
  }

  const int wave = threadIdx.x >> 5;
  const int lane = threadIdx.x & 31;
#pragma unroll
  for (int off = 16; off > 0; off >>= 1) {
    cnt += __shfl_down(cnt, off, 32);
    sx  += __shfl_down(sx, off, 32);
    sy  += __shfl_down(sy, off, 32);
  }
  if (lane == 0) { red[0][wave] = cnt; red[1][wave] = sx; red[2][wave] = sy; }
  __syncthreads();
  if (threadIdx.x == 0) {
    float c = 0.0f, a = 0.0f, d = 0.0f;
#pragma unroll
    for (int w = 0; w < 8; ++w) { c += red[0][w]; a += red[1][w]; d += red[2][w]; }
    res[pid * 3 + 0] = c;
    res[pid * 3 + 1] = a;
    res[pid * 3 + 2] = d;
  }
}

// ---------------- Kernel C: deterministic fixed-order combine ----------------
__global__ __launch_bounds__(256) void finish_kernel(const float* __restrict__ partial,
                                                     const float* __restrict__ res,
                                                     float* __restrict__ out) {
  __shared__ float acc[PAIRS];
  const int t = threadIdx.x;
  if (t < PAIRS) {
    float m = 0.0f, zh = 0.0f;
    for (int s = 0; s < NSLICE; ++s) {
      m  += partial[(t * NSLICE + s) * 2 + 0];
      zh += partial[(t * NSLICE + s) * 2 + 1];
    }
    const float bce = (m - zh) * (1.0f / (float)NPIX);
    const float cnt = res[t * 3 + 0];
    const float sx  = res[t * 3 + 1];
    const float sy  = res[t * 3 + 2];
    acc[t] = 2.0f * bce + sx / cnt + sy / cnt;
  }
  __syncthreads();
  if (t == 0) {
    float s = 0.0f;
    for (int k = 0; k < PAIRS; ++k) s += acc[k];
    out[0] = s / (float)PAIRS;
  }
}

extern "C" void kernel_launch(void* const* d_in, const int* in_sizes, int n_in,
                              void* d_out, int out_size, void* d_ws, size_t ws_size,
                              hipStream_t stream) {
  const float* fm = (const float*)d_in[0];   // (8, 57, 512, 512) fp32
  const float* lm = (const float*)d_in[1];   // (8, 19, 2) fp32
  float* out = (float*)d_out;                // scalar fp32

  float* partial = (float*)d_ws;                       // PAIRS*NSLICE*2 floats = 38912 B
  float* res     = (float*)((char*)d_ws + 40960);      // PAIRS*3 floats

  dim3 gA(PAIRS, NSLICE);
  bce_stream<<<gA, 256, 0, stream>>>(fm, lm, partial);
  l1_disc<<<PAIRS, 256, 0, stream>>>(fm, lm, res);
  finish_kernel<<<1, 256, 0, stream>>>(partial, res, out);
}